// NodeNetGNN_57140244906530
// MI455X (gfx1250) — compile-verified
//
#include <hip/hip_runtime.h>

#define N_CELL 50000
#define N_NET  10000
#define E_PIN  50000
#define E_NEAR 50000
#define D_NODE 64
#define D_NET  64
#define O_NODE 32
#define O_NET  64
#define GCOL   288   // 8 blocks of 32 (per-pin-dim) + 1 bias block of 32

typedef float v2f __attribute__((ext_vector_type(2)));
typedef float v8f __attribute__((ext_vector_type(8)));

// ---------------- degree / count pass ----------------
__global__ __launch_bounds__(256) void count_deg_kernel(
    const int* __restrict__ pins_src, const int* __restrict__ pins_dst,
    const int* __restrict__ pinned_dst, const int* __restrict__ near_dst,
    float* deg_src, float* deg_net, float* cnt_pin, float* cnt_near) {
  int e = blockIdx.x * blockDim.x + threadIdx.x;
  if (e >= E_PIN) return;
  atomicAdd(&deg_src[pins_src[e]], 1.0f);
  atomicAdd(&deg_net[pins_dst[e]], 1.0f);
  atomicAdd(&cnt_pin[pinned_dst[e]], 1.0f);
  atomicAdd(&cnt_near[near_dst[e]], 1.0f);
}

__global__ __launch_bounds__(256) void inv_sqrt_kernel(float* v, int n) {
  int i = blockIdx.x * blockDim.x + threadIdx.x;
  if (i < n) v[i] = rsqrtf(fmaxf(v[i], 1.0f));
}

// Transposed 64x64 weight: out[n*64+k] = in[k*64+n]
__global__ __launch_bounds__(256) void transpose64_kernel(
    const float* __restrict__ in, float* __restrict__ out) {
  int idx = blockIdx.x * blockDim.x + threadIdx.x;   // 4096
  if (idx >= 64 * 64) return;
  int n = idx >> 6, k = idx & 63;
  out[idx] = in[k * 64 + n];
}

// Build BT[288 x 64] (K-contiguous): BT[p*32+o][i] = We[p, i*32+o] (p<8), BT[256+o][i] = be[i*32+o]
__global__ __launch_bounds__(256) void build_edge_matT_kernel(
    const float* __restrict__ We, const float* __restrict__ be,
    float* __restrict__ out) {
  int idx = blockIdx.x * blockDim.x + threadIdx.x;   // 288*64
  if (idx >= GCOL * 64) return;
  int c = idx >> 6, i = idx & 63;
  int p = c >> 5, o = c & 31;
  out[idx] = (p < 8) ? We[p * (D_NET * O_NODE) + i * 32 + o] : be[i * 32 + o];
}

// ---------------- WMMA f32 GEMM: C[MxN] = diag(rowscale) * A[Mx64] @ B[64xN] ----------------
// B supplied TRANSPOSED as BT[N x 64] so each lane's K-pair is one b64 load.
// N compile-time; NB 16-col tiles per wave (A-fragment reuse). 8 waves/block.
template <int N, int NB, bool SCALE>
__global__ __launch_bounds__(256) void gemm_k64_wmma_kernel(
    const float* __restrict__ A, const float* __restrict__ BT,
    float* __restrict__ C, const float* __restrict__ rowscale, int M) {
  constexpr int TG = (N / 16) / NB;          // tile-groups along N (exact)
  const int wave = threadIdx.x >> 5;
  const int lane = threadIdx.x & 31;
  const int tile = blockIdx.x * 8 + wave;
  const int tm = tile / TG;
  if (tm * 16 >= M) return;                  // wave-uniform exit; EXEC stays all-ones
  const int tg = tile - tm * TG;
  const int half = lane >> 4;                // selects K-pair {0,1} vs {2,3}
  const int l16  = lane & 15;

  const float* arow = A + (size_t)(tm * 16 + l16) * 64;
  float s = 1.0f;
  if (SCALE) s = rowscale[tm * 16 + l16];
  const float* bcol = BT + (size_t)(tg * (NB * 16) + l16) * 64;  // lane's column, K-contiguous

  v8f acc[NB];
#pragma unroll
  for (int nb = 0; nb < NB; ++nb) acc[nb] = v8f{0.f, 0.f, 0.f, 0.f, 0.f, 0.f, 0.f, 0.f};

#pragma unroll
  for (int k0 = 0; k0 < 64; k0 += 4) {
    const int ka = k0 + half * 2;            // even -> 8B aligned vector loads
    v2f a = *(const v2f*)(arow + ka);
    if (SCALE) { a.x *= s; a.y *= s; }
#pragma unroll
    for (int nb = 0; nb < NB; ++nb) {
      v2f b = *(const v2f*)(bcol + nb * 16 * 64 + ka);   // constant offset, one b64
      acc[nb] = __builtin_amdgcn_wmma_f32_16x16x4_f32(
          /*neg_a=*/false, a, /*neg_b=*/false, b,
          /*c_mod=*/(short)0, acc[nb], /*reuse_a=*/false, /*reuse_b=*/false);
    }
  }

  float* crow = C + (size_t)(tm * 16 + 8 * half) * N + tg * (NB * 16) + l16;
#pragma unroll
  for (int nb = 0; nb < NB; ++nb)
#pragma unroll
    for (int v = 0; v < 8; ++v)
      crow[(size_t)v * N + nb * 16] = acc[nb][v];        // row = tm*16 + v + 8*half
}

// ---------------- pins scatter: agg[dst[e]][c] += H1[src[e]][c], c in [0,64) ----------------
__global__ __launch_bounds__(256) void pins_scatter_kernel(
    const int* __restrict__ src, const int* __restrict__ dst,
    const float* __restrict__ H1, float* __restrict__ agg) {
  int t = blockIdx.x * blockDim.x + threadIdx.x;
  int e = t >> 6, c = t & 63;
  if (e >= E_PIN) return;
  atomicAdd(&agg[(size_t)dst[e] * O_NET + c], H1[(size_t)src[e] * O_NET + c]);
}

// ---------------- NNConv scatter: msg[o] = sum_p ef[p]*G[src][p*32+o] + G[src][256+o] ----------------
__global__ __launch_bounds__(256) void nnconv_scatter_kernel(
    const int* __restrict__ src, const int* __restrict__ dst,
    const float* __restrict__ efeat, const float* __restrict__ G,
    float* __restrict__ sum, int nedges) {
  int t = blockIdx.x * blockDim.x + threadIdx.x;
  int e = t >> 5, o = t & 31;
  if (e >= nedges) return;
  const float* g = G + (size_t)src[e] * GCOL + o;
  const float* f = efeat + (size_t)e * 8;
  float m = g[256];
#pragma unroll
  for (int p = 0; p < 8; ++p) m += f[p] * g[p * 32];
  atomicAdd(&sum[(size_t)dst[e] * O_NODE + o], m);
}

// ---------------- finalize ----------------
__global__ __launch_bounds__(256) void finalize_cell_kernel(
    const float* __restrict__ sum_p, const float* __restrict__ cnt_p,
    const float* __restrict__ b_pinned,
    const float* __restrict__ sum_n, const float* __restrict__ cnt_n,
    const float* __restrict__ b_near, float* __restrict__ out) {
  int t = blockIdx.x * blockDim.x + threadIdx.x;
  int i = t >> 5, o = t & 31;
  if (i >= N_CELL) return;
  float hp = sum_p[(size_t)i * O_NODE + o] / fmaxf(cnt_p[i], 1.0f) + b_pinned[o];
  float hn = sum_n[(size_t)i * O_NODE + o] / fmaxf(cnt_n[i], 1.0f) + b_near[o];
  out[(size_t)i * O_NODE + o] = fmaxf(hp, hn);
}

__global__ __launch_bounds__(256) void finalize_net_kernel(
    const float* __restrict__ agg, const float* __restrict__ isq_deg_net,
    const float* __restrict__ b_pins, const float* __restrict__ netlin,
    const float* __restrict__ b_net, float* __restrict__ out) {
  int t = blockIdx.x * blockDim.x + threadIdx.x;
  int i = t >> 6, c = t & 63;
  if (i >= N_NET) return;
  out[(size_t)i * O_NET + c] =
      agg[(size_t)i * O_NET + c] * isq_deg_net[i] + b_pins[c] +
      netlin[(size_t)i * O_NET + c] + b_net[c];
}

extern "C" void kernel_launch(void* const* d_in, const int* in_sizes, int n_in,
                              void* d_out, int out_size, void* d_ws, size_t ws_size,
                              hipStream_t stream) {
  (void)in_sizes; (void)n_in; (void)out_size; (void)ws_size;
  const float* node_feat  = (const float*)d_in[0];
  const float* net_feat   = (const float*)d_in[1];
  const float* pin_feat   = (const float*)d_in[2];
  const float* edge_feat  = (const float*)d_in[3];
  const int*   pins_src   = (const int*)d_in[4];
  const int*   pins_dst   = (const int*)d_in[5];
  const int*   pinned_src = (const int*)d_in[6];
  const int*   pinned_dst = (const int*)d_in[7];
  const int*   near_src   = (const int*)d_in[8];
  const int*   near_dst   = (const int*)d_in[9];
  const float* W_pins   = (const float*)d_in[10];
  const float* b_pins   = (const float*)d_in[11];
  const float* W_topo   = (const float*)d_in[12];
  const float* b_topo   = (const float*)d_in[13];
  const float* b_pinned = (const float*)d_in[14];
  const float* W_geom   = (const float*)d_in[15];
  const float* b_geom   = (const float*)d_in[16];
  const float* b_near   = (const float*)d_in[17];
  const float* W_net    = (const float*)d_in[18];
  const float* b_net    = (const float*)d_in[19];

  float* ws = (float*)d_ws;
  size_t o = 0;
  float* deg_src = ws + o; o += N_CELL;                     // pins out-degree (cells)
  float* deg_net = ws + o; o += N_NET;                      // pins in-degree (nets)
  float* cnt_pin = ws + o; o += N_CELL;                     // pinned in-count (cells)
  float* cnt_nr  = ws + o; o += N_CELL;                     // near in-count (cells)
  float* agg_net = ws + o; o += (size_t)N_NET * O_NET;      // pins scatter accum
  float* sum_pin = ws + o; o += (size_t)N_CELL * O_NODE;    // pinned msg accum
  float* sum_nr  = ws + o; o += (size_t)N_CELL * O_NODE;    // near msg accum
  const size_t zero_floats = o;                             // everything above needs zeroing
  float* BtopoT  = ws + o; o += (size_t)GCOL * 64;          // transposed (K-contiguous)
  float* BgeomT  = ws + o; o += (size_t)GCOL * 64;
  float* WpinsT  = ws + o; o += (size_t)64 * 64;
  float* WnetT   = ws + o; o += (size_t)64 * 64;
  float* H1      = ws + o; o += (size_t)N_CELL * O_NET;     // scaled node_feat @ W_pins
  float* Gpin    = ws + o; o += (size_t)N_NET * GCOL;       // net_feat @ Btopo
  float* Gnear   = ws + o; o += (size_t)N_CELL * GCOL;      // node_feat @ Bgeom
  float* netlin  = ws + o; o += (size_t)N_NET * D_NET;      // net_feat @ W_net

  hipMemsetAsync(ws, 0, zero_floats * sizeof(float), stream);

  count_deg_kernel<<<(E_PIN + 255) / 256, 256, 0, stream>>>(
      pins_src, pins_dst, pinned_dst, near_dst, deg_src, deg_net, cnt_pin, cnt_nr);
  inv_sqrt_kernel<<<(N_CELL + 255) / 256, 256, 0, stream>>>(deg_src, N_CELL);
  inv_sqrt_kernel<<<(N_NET + 255) / 256, 256, 0, stream>>>(deg_net, N_NET);

  transpose64_kernel<<<(64 * 64 + 255) / 256, 256, 0, stream>>>(W_pins, WpinsT);
  transpose64_kernel<<<(64 * 64 + 255) / 256, 256, 0, stream>>>(W_net, WnetT);
  build_edge_matT_kernel<<<(GCOL * 64 + 255) / 256, 256, 0, stream>>>(W_topo, b_topo, BtopoT);
  build_edge_matT_kernel<<<(GCOL * 64 + 255) / 256, 256, 0, stream>>>(W_geom, b_geom, BgeomT);

  // WMMA GEMMs. tiles = (M/16) * ((N/16)/NB); NB=2 -> each wave does a 16x32 strip.
  {
    int tiles;
    tiles = (N_CELL / 16) * ((O_NET / 16) / 2);
    gemm_k64_wmma_kernel<O_NET, 2, true><<<(tiles + 7) / 8, 256, 0, stream>>>(
        node_feat, WpinsT, H1, deg_src, N_CELL);
    tiles = (N_CELL / 16) * ((GCOL / 16) / 2);
    gemm_k64_wmma_kernel<GCOL, 2, false><<<(tiles + 7) / 8, 256, 0, stream>>>(
        node_feat, BgeomT, Gnear, nullptr, N_CELL);
    tiles = (N_NET / 16) * ((GCOL / 16) / 2);
    gemm_k64_wmma_kernel<GCOL, 2, false><<<(tiles + 7) / 8, 256, 0, stream>>>(
        net_feat, BtopoT, Gpin, nullptr, N_NET);
    tiles = (N_NET / 16) * ((D_NET / 16) / 2);
    gemm_k64_wmma_kernel<D_NET, 2, false><<<(tiles + 7) / 8, 256, 0, stream>>>(
        net_feat, WnetT, netlin, nullptr, N_NET);
  }

  pins_scatter_kernel<<<((size_t)E_PIN * 64 + 255) / 256, 256, 0, stream>>>(
      pins_src, pins_dst, H1, agg_net);
  nnconv_scatter_kernel<<<((size_t)E_PIN * 32 + 255) / 256, 256, 0, stream>>>(
      pinned_src, pinned_dst, pin_feat, Gpin, sum_pin, E_PIN);
  nnconv_scatter_kernel<<<((size_t)E_NEAR * 32 + 255) / 256, 256, 0, stream>>>(
      near_src, near_dst, edge_feat, Gnear, sum_nr, E_NEAR);

  float* out_cell = (float*)d_out;                          // (50000, 32)
  float* out_net  = out_cell + (size_t)N_CELL * O_NODE;     // (10000, 64)
  finalize_cell_kernel<<<((size_t)N_CELL * 32 + 255) / 256, 256, 0, stream>>>(
      sum_pin, cnt_pin, b_pinned, sum_nr, cnt_nr, b_near, out_cell);
  finalize_net_kernel<<<((size_t)N_NET * 64 + 255) / 256, 256, 0, stream>>>(
      agg_net, deg_net, b_pins, netlin, b_net, out_net);
}